// GramSchmidt_47175920779340
// MI455X (gfx1250) — compile-verified
//
#include <hip/hip_runtime.h>

// CDNA5 (gfx1250) fused Gram-Schmidt-variant kernel.
// Out = V * A^T where A comes from a 64-step scalar recurrence on G = V^T V.
// All matrix work in V_WMMA_F32_16X16X4_F32 (full f32 precision, matrix pipe).

typedef __attribute__((ext_vector_type(2))) float v2f;
typedef __attribute__((ext_vector_type(8))) float v8f;

#define WMMA_F32(a, b, c) \
  __builtin_amdgcn_wmma_f32_16x16x4_f32(false, (a), false, (b), (short)0, (c), false, false)

__launch_bounds__(128)
__global__ void gs_fused_kernel(const float* __restrict__ y, float* __restrict__ out) {
  __shared__ float Gs[64 * 64];  // Gram matrix G = V^T V
  __shared__ float As[64 * 64];  // coefficient matrix, row i = a_i

  const int b = blockIdx.x;
  const float* V = y + (size_t)b * 512 * 64;   // 512 rows (R then I), 64 cols
  float* O = out + (size_t)b * 512 * 64;

  const int tid  = threadIdx.x;
  const int wave = tid >> 5;      // 4 waves
  const int lane = tid & 31;      // wave32
  const int half = lane >> 4;     // K-pair selector in 16x4 frags
  const int mr   = lane & 15;     // row/col within 16-tile

  // ---------------- Phase A: G = V^T V  (64x64, K=512) ----------------
  // WMMA 16x16x4 f32. A-frag (16x4 of V^T) and B-frag (4x16 of V) have the
  // SAME lane layout here: elem = V[k0 + 2*half + {0,1}][16*t + mr].
  {
    v8f acc0 = {}, acc1 = {}, acc2 = {}, acc3 = {};
    for (int k0 = 0; k0 < 512; k0 += 4) {
      const float* p0 = V + (size_t)(k0 + 2 * half) * 64 + mr;
      const float* p1 = p0 + 64;
      v2f af; af.x = p0[16 * wave]; af.y = p1[16 * wave];  // this wave's tile strip
      v2f b0; b0.x = p0[0];  b0.y = p1[0];
      v2f b1; b1.x = p0[16]; b1.y = p1[16];
      v2f b2; b2.x = p0[32]; b2.y = p1[32];
      v2f b3; b3.x = p0[48]; b3.y = p1[48];
      acc0 = WMMA_F32(af, b0, acc0);
      acc1 = WMMA_F32(af, b1, acc1);
      acc2 = WMMA_F32(af, b2, acc2);
      acc3 = WMMA_F32(af, b3, acc3);
    }
#pragma unroll
    for (int r = 0; r < 8; ++r) {  // C layout: VGPR r -> M = r + 8*half, N = mr
      int row = 16 * wave + r + 8 * half;
      Gs[row * 64 +  0 + mr] = acc0[r];
      Gs[row * 64 + 16 + mr] = acc1[r];
      Gs[row * 64 + 32 + mr] = acc2[r];
      Gs[row * 64 + 48 + mr] = acc3[r];
    }
  }
  __syncthreads();

  // ---------------- Phase B: 64-step recurrence -> A (wave 0) ----------------
  // a_i = (e_i - s_i a_{i-1}) / d_i ;  u_i = (G[:,i] - s_i u_{i-1}) / d_i
  // g_i = (u_{i-1}[i] - s_i p_{i-1}) / d_i ; p_i = (u_i[i] - s_i g_i) / d_i
  // chain regs: R_t <- u_i[t] - R_t * g_i ; s_i = R_i at start of step i.
  if (wave == 0) {
    float aP0 = 0.f, aP1 = 0.f;  // a_{i-1}[lane], a_{i-1}[lane+32]
    float uP0 = 0.f, uP1 = 0.f;  // u_{i-1}
    float R0 = 0.f, R1 = 0.f;    // running chain values
    float p = 0.f;
    for (int i = 0; i < 64; ++i) {
      float g0 = Gs[i * 64 + lane];        // G row i == G col i (symmetric)
      float g1 = Gs[i * 64 + lane + 32];
      float dlo = __shfl(g0, i & 31, 32);
      float dhi = __shfl(g1, i & 31, 32);
      float di = (i < 32) ? dlo : dhi;     // d_i = G[i][i]
      float inv_d = 1.0f / di;
      float s = 0.f;
      if (i > 0) {
        float slo = __shfl(R0, i & 31, 32);
        float shi = __shfl(R1, i & 31, 32);
        s = (i < 32) ? slo : shi;          // s_i
      }
      aP0 = (((lane == i) ? 1.f : 0.f) - s * aP0) * inv_d;
      aP1 = (((lane + 32 == i) ? 1.f : 0.f) - s * aP1) * inv_d;
      float nu0 = (g0 - s * uP0) * inv_d;
      float nu1 = (g1 - s * uP1) * inv_d;
      if (i == 0) {
        p = __shfl(nu0, 0, 32) * inv_d;    // p_0 = u_0[0] / d_0
        R0 = nu0; R1 = nu1;                // chain init: R_t = u_0[t]
      } else {
        float uplo = __shfl(uP0, i & 31, 32);
        float uphi = __shfl(uP1, i & 31, 32);
        float up_i = (i < 32) ? uplo : uphi;   // u_{i-1}[i]
        float gi = (up_i - s * p) * inv_d;     // <Q_{i-1}, Q_i>
        float nlo = __shfl(nu0, i & 31, 32);
        float nhi = __shfl(nu1, i & 31, 32);
        float nu_i = (i < 32) ? nlo : nhi;     // u_i[i]
        p = (nu_i - s * gi) * inv_d;
        R0 = nu0 - R0 * gi;
        R1 = nu1 - R1 * gi;
      }
      uP0 = nu0; uP1 = nu1;
      As[i * 64 + lane]      = aP0;
      As[i * 64 + lane + 32] = aP1;
    }
  }
  __syncthreads();

  // ---------------- Phase C: Out = V * A^T  (512x64 x 64x64) ----------------
  // A-frag: V[m0+mr][k0 + 2*half + {0,1}] (contiguous pair -> b64 load).
  // B-frag (of A^T): As[16*tn + mr][k0 + 2*half + {0,1}] from LDS.
  for (int q = 0; q < 8; ++q) {
    const int m0 = 16 * (wave + 4 * q);
    v8f c0 = {}, c1 = {}, c2 = {}, c3 = {};
    for (int k0 = 0; k0 < 64; k0 += 4) {
      const float* pa = V + (size_t)(m0 + mr) * 64 + k0 + 2 * half;
      v2f af; af.x = pa[0]; af.y = pa[1];
      const float* pb = As + mr * 64 + k0 + 2 * half;
      v2f b0; b0.x = pb[0];           b0.y = pb[1];
      v2f b1; b1.x = pb[16 * 64];     b1.y = pb[16 * 64 + 1];
      v2f b2; b2.x = pb[32 * 64];     b2.y = pb[32 * 64 + 1];
      v2f b3; b3.x = pb[48 * 64];     b3.y = pb[48 * 64 + 1];
      c0 = WMMA_F32(af, b0, c0);
      c1 = WMMA_F32(af, b1, c1);
      c2 = WMMA_F32(af, b2, c2);
      c3 = WMMA_F32(af, b3, c3);
    }
#pragma unroll
    for (int r = 0; r < 8; ++r) {
      int row = m0 + r + 8 * half;
      O[(size_t)row * 64 +  0 + mr] = c0[r];
      O[(size_t)row * 64 + 16 + mr] = c1[r];
      O[(size_t)row * 64 + 32 + mr] = c2[r];
      O[(size_t)row * 64 + 48 + mr] = c3[r];
    }
  }
}

extern "C" void kernel_launch(void* const* d_in, const int* in_sizes, int n_in,
                              void* d_out, int out_size, void* d_ws, size_t ws_size,
                              hipStream_t stream) {
  (void)n_in; (void)out_size; (void)d_ws; (void)ws_size;
  const float* y = (const float*)d_in[0];
  float* out = (float*)d_out;
  const int B = in_sizes[0] / (512 * 64);  // 2048 for the reference shapes
  hipLaunchKernelGGL(gs_fused_kernel, dim3(B), dim3(128), 0, stream, y, out);
}